// IndoBERT_CRF_30064771072202
// MI455X (gfx1250) — compile-verified
//
#include <hip/hip_runtime.h>
#include <cstdint>
#include <cstddef>

typedef float v2f __attribute__((ext_vector_type(2)));
typedef float v8f __attribute__((ext_vector_type(8)));
typedef unsigned int u32x4 __attribute__((ext_vector_type(4)));
typedef int i32x4 __attribute__((ext_vector_type(4)));
typedef int i32x8 __attribute__((ext_vector_type(8)));

#define B_   64
#define S_   512
#define W_   384
#define D_   768
#define T_   21
#define EMS  32                       // padded emission row stride (floats)
#define KCHUNK 64
#define NCHUNK (D_ / KCHUNK)          // 12
#define WAVES  8

// workspace layout (in floats)
#define WT_OFF  0                     // padded W_tag: 32 x 768
#define EM_OFF  (32 * D_)             // emissions: 24576 x 32
#define LLH_OFF (EM_OFF + (B_ * W_) * EMS)  // llh: 64

#if __has_builtin(__builtin_amdgcn_tensor_load_to_lds) && \
    __has_builtin(__builtin_amdgcn_s_wait_tensorcnt)
#define USE_TDM 1
#else
#define USE_TDM 0
#endif

// ---------------------------------------------------------------------------
// Phase 0: zero-pad W_tag (21x768) into a 32x768 WMMA B-matrix in workspace
// ---------------------------------------------------------------------------
__global__ void pad_kernel(const float* __restrict__ W_tag,
                           float* __restrict__ wtpad) {
  int i = blockIdx.x * blockDim.x + threadIdx.x;
  int stride = gridDim.x * blockDim.x;
  for (int j = i; j < 32 * D_; j += stride) {
    int n = j / D_;
    int k = j - n * D_;
    wtpad[j] = (n < T_) ? W_tag[n * D_ + k] : 0.0f;
  }
}

// ---------------------------------------------------------------------------
// Phase 1: fused gather + emissions GEMM.
//   A tiles: TDM gather-mode DMA (16 sorted row indices) -> LDS, double
//            buffered, tracked with TENSORcnt. B: L2-resident padded W_tag.
//   Math:    V_WMMA_F32_16X16X4_F32, two N-tiles (0..15, 16..20).
// grid = (B*W)/(WAVES*16) = 192 blocks, 256 threads (8 waves, 1 M-tile each)
// ---------------------------------------------------------------------------
__global__ __launch_bounds__(256)
void emis_kernel(const float* __restrict__ hidden,
                 const float* __restrict__ wtpad,
                 const float* __restrict__ b_tag,
                 const int*   __restrict__ word_maps,
                 float*       __restrict__ em) {
  __shared__ float Alds[WAVES][2][16][KCHUNK];   // 64 KB, double buffered

  const int lane = threadIdx.x & 31;
  const int wave = threadIdx.x >> 5;
  const int hi   = lane >> 4;
  const int lo   = lane & 15;
  const int rowbase = (blockIdx.x * WAVES + wave) * 16;
  const int b = rowbase / W_;          // 16-row tiles never straddle a batch

  v8f c0 = {};   // N = 0..15
  v8f c1 = {};   // N = 16..31 (only 16..20 stored)

#if USE_TDM
  // ---- pack the 16 sorted gather indices (16-bit each) into D# groups 2/3
  int idxw[8];
  #pragma unroll
  for (int i = 0; i < 8; ++i) {
    int s0 = word_maps[rowbase + 2 * i];
    int s1 = word_maps[rowbase + 2 * i + 1];
    idxw[i] = (s0 & 0xFFFF) | (s1 << 16);
  }
  const i32x4 g2 = { idxw[0], idxw[1], idxw[2], idxw[3] };
  const i32x4 g3 = { idxw[4], idxw[5], idxw[6], idxw[7] };
  const i32x8 g4 = { 0, 0, 0, 0, 0, 0, 0, 0 };     // zero-filled trailing group

  // ---- D# group 1 (constant across chunks)
  i32x8 g1;
  g1[0] = (2 << 16);                               // data_size = 4 bytes
  g1[1] = (D_ & 0xFFFF) << 16;                     // tensor_dim0 = 768 (lo16)
  g1[2] = ((D_ >> 16) & 0xFFFF) | ((S_ & 0xFFFF) << 16); // dim0 hi16, tensor_dim1 = 512
  g1[3] = ((S_ >> 16) & 0xFFFF) | (KCHUNK << 16);  // dim1 hi16, tile_dim0 = 64
  g1[4] = 16;                                      // tile_dim1 = 16 valid indices
  g1[5] = D_;                                      // tensor_dim0_stride = 768 (lo32)
  g1[6] = 0;
  g1[7] = 0;

  const uint64_t gbase = (uint64_t)(uintptr_t)hidden +
                         (uint64_t)b * S_ * D_ * sizeof(float);

  auto issue_tdm = [&](int buf, int k0) {
    uint32_t lds = (uint32_t)(uintptr_t)&Alds[wave][buf][0][0];
    uint64_t ga  = gbase + (uint64_t)k0 * sizeof(float);
    u32x4 g0;
    g0[0] = 1u | (1u << 31);                       // count=1, gather_mode=1, 16b idx
    g0[1] = lds;                                   // lds_addr
    g0[2] = (uint32_t)ga;                          // global_addr[31:0]
    g0[3] = (uint32_t)((ga >> 32) & 0x01FFFFFFu)   // global_addr[56:32]
            | (2u << 30);                          // type = 2 ("image")
    __builtin_amdgcn_tensor_load_to_lds(g0, g1, g2, g3, g4, 0);
  };

  issue_tdm(0, 0);
  for (int c = 0; c < NCHUNK; ++c) {
    const int k0 = c * KCHUNK;
    if (c + 1 < NCHUNK) {
      issue_tdm((c + 1) & 1, k0 + KCHUNK);
      __builtin_amdgcn_s_wait_tensorcnt(1);        // chunk c landed, c+1 in flight
    } else {
      __builtin_amdgcn_s_wait_tensorcnt(0);        // last chunk: drain
    }
    asm volatile("" ::: "memory");                 // order LDS reads after wait
    const float* Ab = &Alds[wave][c & 1][0][0];
    #pragma unroll
    for (int kk = 0; kk < KCHUNK / 4; ++kk) {
      v2f a = *(const v2f*)&Ab[lo * KCHUNK + 4 * kk + 2 * hi];
      const float* bp = wtpad + lo * D_ + (k0 + 4 * kk + 2 * hi);
      v2f b0 = *(const v2f*)bp;
      v2f b1 = *(const v2f*)(bp + 16 * D_);
      c0 = __builtin_amdgcn_wmma_f32_16x16x4_f32(false, a, false, b0,
                                                 (short)0, c0, false, false);
      c1 = __builtin_amdgcn_wmma_f32_16x16x4_f32(false, a, false, b1,
                                                 (short)0, c1, false, false);
    }
    asm volatile("" ::: "memory");                 // buffer consumed before reuse
  }
#else
  // ---- fallback: manual coalesced staging ----
  for (int c = 0; c < NCHUNK; ++c) {
    const int k0 = c * KCHUNK;
    for (int r = lane >> 4; r < 16; r += 2) {
      int s = word_maps[rowbase + r];
      const float4* src = (const float4*)(hidden + (size_t)(b * S_ + s) * D_ + k0);
      *(float4*)&Alds[wave][0][r][4 * lo] = src[lo];
    }
    __syncthreads();
    #pragma unroll
    for (int kk = 0; kk < KCHUNK / 4; ++kk) {
      v2f a = *(const v2f*)&Alds[wave][0][lo][4 * kk + 2 * hi];
      const float* bp = wtpad + lo * D_ + (k0 + 4 * kk + 2 * hi);
      v2f b0 = *(const v2f*)bp;
      v2f b1 = *(const v2f*)(bp + 16 * D_);
      c0 = __builtin_amdgcn_wmma_f32_16x16x4_f32(false, a, false, b0,
                                                 (short)0, c0, false, false);
      c1 = __builtin_amdgcn_wmma_f32_16x16x4_f32(false, a, false, b1,
                                                 (short)0, c1, false, false);
    }
    __syncthreads();
  }
#endif

  // ---- add bias, store padded emissions ----
  float bt0 = b_tag[lo];
  int   n1  = 16 + lo;
  float bt1 = (n1 < T_) ? b_tag[n1] : 0.0f;
  #pragma unroll
  for (int v = 0; v < 8; ++v) {
    int row = rowbase + v + 8 * hi;   // C/D layout: M = v + 8*(lane>>4)
    em[(size_t)row * EMS + lo] = c0[v] + bt0;
    if (n1 < T_) em[(size_t)row * EMS + n1] = c1[v] + bt1;
  }
}

// ---------------------------------------------------------------------------
// Phase 2: CRF numerator + forward logsumexp scan. One wave32 per batch.
// ---------------------------------------------------------------------------
__global__ __launch_bounds__(32)
void crf_kernel(const float* __restrict__ em,
                const float* __restrict__ trans,
                const float* __restrict__ startT,
                const float* __restrict__ endT,
                const int*   __restrict__ tags,
                const unsigned char* __restrict__ mask,
                float* __restrict__ llh) {
  const int   b    = blockIdx.x;
  const int   lane = threadIdx.x;
  const bool  act  = lane < T_;
  const int   tcl  = act ? lane : 0;
  const float NEG  = -1e30f;

  // exp(trans[:, t]) hoisted into registers (column per lane)
  float Et[T_];
  #pragma unroll
  for (int p = 0; p < T_; ++p) Et[p] = __expf(trans[p * T_ + tcl]);

  const float* emb = em + (size_t)(b * W_) * EMS;
  float score = act ? (startT[lane] + emb[lane]) : NEG;

  for (int w = 1; w < W_; ++w) {
    bool m = mask[b * W_ + w] != 0;
    float M = score;
    #pragma unroll
    for (int off = 16; off; off >>= 1) M = fmaxf(M, __shfl_xor(M, off, 32));
    float a = __expf(score - M);       // inactive lanes: exp(-inf) = 0
    float acc = 0.0f;
    #pragma unroll
    for (int p = 0; p < T_; ++p)
      acc = fmaf(__shfl(a, p, 32), Et[p], acc);
    float ns = emb[(size_t)w * EMS + tcl] + M + __logf(acc);
    if (act && m) score = ns;
  }

  // log_z = logsumexp(score + end_trans)
  float v  = act ? (score + endT[lane]) : NEG;
  float M2 = v;
  #pragma unroll
  for (int off = 16; off; off >>= 1) M2 = fmaxf(M2, __shfl_xor(M2, off, 32));
  float s = act ? __expf(v - M2) : 0.0f;
  #pragma unroll
  for (int off = 16; off; off >>= 1) s += __shfl_xor(s, off, 32);
  float log_z = M2 + __logf(s);

  // numerator (parallel over w, deterministic wave reduction)
  int   cnt = 0;
  float numpart = 0.0f;
  for (int w = lane; w < W_; w += 32) {
    bool m = mask[b * W_ + w] != 0;
    cnt += m ? 1 : 0;
    if (w == 0) {
      int t0 = tags[b * W_];
      numpart += startT[t0] + emb[t0];
    } else if (m) {
      int tp = tags[b * W_ + w - 1];
      int tn = tags[b * W_ + w];
      numpart += trans[tp * T_ + tn] + emb[(size_t)w * EMS + tn];
    }
  }
  #pragma unroll
  for (int off = 16; off; off >>= 1) {
    numpart += __shfl_xor(numpart, off, 32);
    cnt     += __shfl_xor(cnt, off, 32);
  }
  if (lane == 0) {
    int last  = tags[b * W_ + cnt - 1];
    float num = numpart + endT[last];
    llh[b] = num - log_z;
  }
}

// ---------------------------------------------------------------------------
// Phase 3: deterministic mean over 64 batches -> scalar output
// ---------------------------------------------------------------------------
__global__ __launch_bounds__(32)
void finish_kernel(const float* __restrict__ llh, float* __restrict__ out) {
  int lane = threadIdx.x;
  float v = llh[lane] + llh[lane + 32];
  #pragma unroll
  for (int off = 16; off; off >>= 1) v += __shfl_xor(v, off, 32);
  if (lane == 0) *out = -v / (float)B_;
}

// ---------------------------------------------------------------------------
extern "C" void kernel_launch(void* const* d_in, const int* in_sizes, int n_in,
                              void* d_out, int out_size, void* d_ws, size_t ws_size,
                              hipStream_t stream) {
  const float* hidden    = (const float*)d_in[0];
  const float* W_tag     = (const float*)d_in[1];
  const float* b_tag     = (const float*)d_in[2];
  const float* startT    = (const float*)d_in[3];
  const float* endT      = (const float*)d_in[4];
  const float* trans     = (const float*)d_in[5];
  const int*   word_maps = (const int*)d_in[6];
  const int*   tags      = (const int*)d_in[7];
  const unsigned char* word_mask = (const unsigned char*)d_in[8];

  float* ws    = (float*)d_ws;
  float* wtpad = ws + WT_OFF;
  float* em    = ws + EM_OFF;
  float* llh   = ws + LLH_OFF;
  float* out   = (float*)d_out;

  pad_kernel<<<32, 256, 0, stream>>>(W_tag, wtpad);
  emis_kernel<<<(B_ * W_) / (WAVES * 16), 256, 0, stream>>>(
      hidden, wtpad, b_tag, word_maps, em);
  crf_kernel<<<B_, 32, 0, stream>>>(em, trans, startT, endT, tags, word_mask, llh);
  finish_kernel<<<1, 32, 0, stream>>>(llh, out);
}